// MagnetLoss_5617817223600
// MI455X (gfx1250) — compile-verified
//
#include <hip/hip_runtime.h>

typedef __attribute__((ext_vector_type(2))) float v2f;
typedef __attribute__((ext_vector_type(4))) float v4f;
typedef __attribute__((ext_vector_type(8))) float v8f;

#define BS 4096
#define DIM 512
#define EPSV 1e-6f
#define MARGIN 2.0f

#define NBY (BS / 128)        // 32 i-blocks (128 rows each)
#define NBX (BS / 64)         // 64 j-blocks (64 cols each)
#define NPART (NBY * NBX)     // 2048 per-block partials

// ---------------------------------------------------------------------------
// Kernel 1: per-row squared norm and row sum. One row per wave (wave32).
// ---------------------------------------------------------------------------
__global__ __launch_bounds__(256) void row_stats_kernel(const float* __restrict__ F,
                                                        float* __restrict__ sqn,
                                                        float* __restrict__ sm) {
    const int lane = threadIdx.x & 31;
    const int wave = threadIdx.x >> 5;
    const int row  = blockIdx.x * 8 + wave;   // 512 blocks * 8 waves = 4096 rows

    const float* rp = F + (size_t)row * DIM;
    float s = 0.0f, sq = 0.0f;
#pragma unroll
    for (int q = 0; q < 4; ++q) {
        v4f v = *(const v4f*)(rp + q * 128 + lane * 4);
        s  += v.x + v.y + v.z + v.w;
        sq += v.x * v.x + v.y * v.y + v.z * v.z + v.w * v.w;
    }
#pragma unroll
    for (int off = 16; off >= 1; off >>= 1) {
        s  += __shfl_xor(s, off, 32);
        sq += __shfl_xor(sq, off, 32);
    }
    if (lane == 0) {
        sqn[row] = sq;
        sm[row]  = s;
    }
}

// ---------------------------------------------------------------------------
// Loss contribution for ordered pair (ii, jj), ii != jj guaranteed by caller.
// ---------------------------------------------------------------------------
__device__ __forceinline__ float contrib(float g,
                                         int li, int lj,
                                         float sqi, float sqj,
                                         float smi, float smj) {
    float sq = sqi + sqj - 2.0f * g
             + 2.0f * EPSV * (smi - smj)
             + (float)DIM * EPSV * EPSV;
    sq = fmaxf(sq, 0.0f);
    if (li == lj) return sq;
    float d = sqrtf(sq);
    float t = fmaxf(MARGIN - d, 0.0f);
    return t * t;
}

// ---------------------------------------------------------------------------
// Kernel 2: fused F*F^T (WMMA f32 16x16x4) + MagnetLoss epilogue.
// Symmetry: only elements with jj > ii contribute; each yields BOTH ordered
// pairs (i,j) and (j,i) from the same Gram value (eps term is antisymmetric).
// Block = 256 threads = 8 waves; block computes a 128x64 super-tile.
// Wave w: rows [wm*32, wm*32+32), cols [wn*32, wn*32+32) -> 2x2 WMMA tiles.
// Blocks entirely below the diagonal exit with a zero partial.
// ---------------------------------------------------------------------------
__global__ __launch_bounds__(256) void gemm_loss_kernel(const float* __restrict__ F,
                                                        const int* __restrict__ labels,
                                                        const float* __restrict__ sqn,
                                                        const float* __restrict__ sm,
                                                        float* __restrict__ partials) {
    __shared__ float As[128][68];  // 68-float row stride: 272B = 16B-aligned rows
    __shared__ float Bs[64][68];
    __shared__ float red[256];

    const int iBase = blockIdx.y * 128;
    const int jBase = blockIdx.x * 64;
    const int tid   = threadIdx.x;
    const int bid   = blockIdx.y * gridDim.x + blockIdx.x;

    if (jBase + 64 <= iBase) {      // entire block in strict lower triangle
        if (tid == 0) partials[bid] = 0.0f;
        return;
    }

    const int lane = tid & 31;
    const int w    = tid >> 5;
    const int wm   = w & 3;         // 0..3 : 32-row group
    const int wn   = w >> 2;        // 0..1 : 32-col group
    const int half = lane >> 4;     // 0/1  : K-pair selector (ISA A/B layout)
    const int m    = lane & 15;     // M (A) / N (B) within tile

    v8f acc00 = {}, acc01 = {}, acc10 = {}, acc11 = {};

    for (int kc = 0; kc < DIM; kc += 64) {
        // Stage A: 128 rows x 64 cols = 2048 float4, 8 per thread.
#pragma unroll
        for (int q = 0; q < 8; ++q) {
            int idx = tid + q * 256;
            int r   = idx >> 4;
            int c   = (idx & 15) * 4;
            *(v4f*)&As[r][c] = *(const v4f*)(F + (size_t)(iBase + r) * DIM + kc + c);
        }
        // Stage B: 64 rows x 64 cols = 1024 float4, 4 per thread.
#pragma unroll
        for (int q = 0; q < 4; ++q) {
            int idx = tid + q * 256;
            int r   = idx >> 4;
            int c   = (idx & 15) * 4;
            *(v4f*)&Bs[r][c] = *(const v4f*)(F + (size_t)(jBase + r) * DIM + kc + c);
        }
        __syncthreads();

        // Prefetch next K chunk (global_prefetch_b8, keep-near locality).
        if (kc + 64 < DIM) {
            __builtin_prefetch(F + (size_t)(iBase + (tid >> 1)) * DIM + kc + 64 + (tid & 1) * 32, 0, 3);
            __builtin_prefetch(F + (size_t)(jBase + (tid >> 2)) * DIM + kc + 64 + (tid & 3) * 16, 0, 3);
        }

#pragma unroll
        for (int kk = 0; kk < 64; kk += 4) {
            const int k = kk + 2 * half;
            // A frags: lane -> M = m, VGPRs hold K = {k, k+1}
            v2f a0 = *(const v2f*)&As[wm * 32 + m][k];
            v2f a1 = *(const v2f*)&As[wm * 32 + 16 + m][k];
            // B frags: lane -> N = m, VGPRs hold K = {k, k+1} (B = F^T tile)
            v2f b0 = *(const v2f*)&Bs[wn * 32 + m][k];
            v2f b1 = *(const v2f*)&Bs[wn * 32 + 16 + m][k];
            acc00 = __builtin_amdgcn_wmma_f32_16x16x4_f32(false, a0, false, b0,
                                                          (short)0, acc00, false, false);
            acc01 = __builtin_amdgcn_wmma_f32_16x16x4_f32(false, a0, false, b1,
                                                          (short)0, acc01, false, false);
            acc10 = __builtin_amdgcn_wmma_f32_16x16x4_f32(false, a1, false, b0,
                                                          (short)0, acc10, false, false);
            acc11 = __builtin_amdgcn_wmma_f32_16x16x4_f32(false, a1, false, b1,
                                                          (short)0, acc11, false, false);
        }
        __syncthreads();
    }

    // ---- Fused epilogue. C/D layout: VGPR v, lane l -> M = v + 8*(l>>4), N = l&15.
    float local = 0.0f;
    const int j0 = jBase + wn * 32 + m;
    const int j1 = j0 + 16;

    const int   lj0 = labels[j0], lj1 = labels[j1];
    const float qj0 = sqn[j0],    qj1 = sqn[j1];
    const float mj0 = sm[j0],     mj1 = sm[j1];

#pragma unroll
    for (int tm = 0; tm < 2; ++tm) {
#pragma unroll
        for (int v = 0; v < 8; ++v) {
            const int   ii = iBase + wm * 32 + tm * 16 + v + 8 * half;
            const int   li = labels[ii];
            const float qi = sqn[ii];
            const float mi = sm[ii];
            const float g0 = tm ? acc10[v] : acc00[v];
            const float g1 = tm ? acc11[v] : acc01[v];
            if (j0 > ii) {
                local += contrib(g0, li, lj0, qi, qj0, mi, mj0)   // (i, j)
                       + contrib(g0, lj0, li, qj0, qi, mj0, mi);  // (j, i)
            }
            if (j1 > ii) {
                local += contrib(g1, li, lj1, qi, qj1, mi, mj1)
                       + contrib(g1, lj1, li, qj1, qi, mj1, mi);
            }
        }
    }

    // Deterministic per-block partial (no float atomics).
    red[tid] = local;
    __syncthreads();
#pragma unroll
    for (int s = 128; s > 0; s >>= 1) {
        if (tid < s) red[tid] += red[tid + s];
        __syncthreads();
    }
    if (tid == 0) partials[bid] = red[0];
}

// ---------------------------------------------------------------------------
// Kernel 3: deterministic final reduction of the 2048 block partials.
// ---------------------------------------------------------------------------
__global__ __launch_bounds__(256) void finalize_kernel(const float* __restrict__ partials,
                                                       float* __restrict__ out) {
    __shared__ float red[256];
    float s = 0.0f;
    for (int i = threadIdx.x; i < NPART; i += 256) s += partials[i];
    red[threadIdx.x] = s;
    __syncthreads();
#pragma unroll
    for (int st = 128; st > 0; st >>= 1) {
        if (threadIdx.x < st) red[threadIdx.x] += red[threadIdx.x + st];
        __syncthreads();
    }
    if (threadIdx.x == 0)
        out[0] = red[0] / ((float)BS * (float)BS - (float)BS);
}

extern "C" void kernel_launch(void* const* d_in, const int* in_sizes, int n_in,
                              void* d_out, int out_size, void* d_ws, size_t ws_size,
                              hipStream_t stream) {
    const float* F      = (const float*)d_in[0];
    const int*   labels = (const int*)d_in[1];
    float*       out    = (float*)d_out;

    float* ws       = (float*)d_ws;
    float* sqn      = ws;                 // 4096 squared norms
    float* smp      = ws + BS;            // 4096 row sums
    float* partials = ws + 2 * BS;        // 2048 block partials

    row_stats_kernel<<<BS / 8, 256, 0, stream>>>(F, sqn, smp);
    gemm_loss_kernel<<<dim3(NBX, NBY), 256, 0, stream>>>(F, labels, sqn, smp, partials);
    finalize_kernel<<<1, 256, 0, stream>>>(partials, out);
}